// SoftTripleLoss_91122026152124
// MI455X (gfx1250) — compile-verified
//
#include <hip/hip_runtime.h>
#include <hip/hip_bf16.h>

typedef float v2f __attribute__((ext_vector_type(2)));
typedef float v8f __attribute__((ext_vector_type(8)));

#define C_TOTAL 10000
#define EMB_D   128
#define KPROX   10
#define LAMBDA_ 20.0f
#define MARGIN  0.01f
#define INV_GAMMA 10.0f   // 1/0.1

// ---------------------------------------------------------------------------
// Pass 1: dots GEMM via V_WMMA_F32_16X16X4_F32, softmax over K=10 proxies,
// pair-average -> S_avg (128 x 10000) in workspace.
//
// Block: 256 threads = 8 waves. blockIdx.x = class octet (8 classes = 80 cols
// = 5 WMMA N-tiles). Wave w handles row-pair tile: rows [16w,16w+16) and
// [16w+128, 16w+144) so S_avg is formed in-register.
// ---------------------------------------------------------------------------
__global__ __launch_bounds__(256) void st_pass1(const float* __restrict__ F,
                                                const float* __restrict__ P,
                                                float* __restrict__ Savg) {
  __shared__ float lds[8][16 * 80];   // per-wave 16x80 dots strip (40 KB total)

  const int warp = threadIdx.x >> 5;
  const int lane = threadIdx.x & 31;
  const int half = lane >> 4;   // selects K sub-pair for A/B, M block for C/D
  const int lp   = lane & 15;   // M (A) / N (B,C,D) index within tile

  const int  m0 = warp * 16;              // rows m0..m0+15 (pair: +128)
  const int  c0 = blockIdx.x * 8;         // 8 classes per block
  const long n0 = (long)c0 * KPROX;       // flattened (class,proxy) column base

  float* tile = lds[warp];

  v8f acc[2][5];
  #pragma unroll
  for (int mt = 0; mt < 2; ++mt)
    #pragma unroll
    for (int j = 0; j < 5; ++j)
      acc[mt][j] = (v8f){0.f, 0.f, 0.f, 0.f, 0.f, 0.f, 0.f, 0.f};

  // K loop over D=128 in steps of 4 (WMMA 16x16x4 f32).
  // A frag layout (32-bit 16x4): lane lp = row M, VGPR0/1 = K {2*half, 2*half+1}.
  // B frag layout (32-bit 4x16): lane lp = col N, VGPR0/1 = K {2*half, 2*half+1}.
  #pragma unroll 2
  for (int d0 = 0; d0 < EMB_D; d0 += 4) {
    v2f a[2], b[5];
    #pragma unroll
    for (int mt = 0; mt < 2; ++mt) {
      const int row = m0 + mt * 128 + lp;
      a[mt] = *(const v2f*)(F + (long)row * EMB_D + d0 + 2 * half);
    }
    #pragma unroll
    for (int j = 0; j < 5; ++j) {
      const long col = n0 + j * 16 + lp;   // row of P viewed as (100000 x 128)
      b[j] = *(const v2f*)(P + col * EMB_D + d0 + 2 * half);
    }
    #pragma unroll
    for (int mt = 0; mt < 2; ++mt)
      #pragma unroll
      for (int j = 0; j < 5; ++j)
        acc[mt][j] = __builtin_amdgcn_wmma_f32_16x16x4_f32(
            false, a[mt], false, b[j], (short)0, acc[mt][j], false, false);
  }

  // For each half of the row pair: spill 16x80 dots to LDS, then each lane
  // handles 4 (row,class) pairs of the 10-wide softmax-weighted sum.
  float Sav[4];
  #pragma unroll
  for (int mt = 0; mt < 2; ++mt) {
    // C/D layout: VGPR vr, lanes 0-15 -> M=vr, lanes 16-31 -> M=vr+8.
    #pragma unroll
    for (int j = 0; j < 5; ++j)
      #pragma unroll
      for (int vr = 0; vr < 8; ++vr)
        tile[(vr + 8 * half) * 80 + j * 16 + lp] = acc[mt][j][vr];

    __builtin_amdgcn_wave_barrier();   // keep DS stores before DS loads

    #pragma unroll
    for (int p = 0; p < 4; ++p) {
      const int idx = lane * 4 + p;    // 0..127 -> (row, class) pair
      const int r   = idx >> 3;        // 0..15
      const int cl  = idx & 7;         // 0..7
      const float* dvec = tile + r * 80 + cl * KPROX;
      float mx = dvec[0];
      #pragma unroll
      for (int k = 1; k < KPROX; ++k) mx = fmaxf(mx, dvec[k]);
      float sw = 0.f, swd = 0.f;
      #pragma unroll
      for (int k = 0; k < KPROX; ++k) {
        const float d = dvec[k];
        const float e = expf((d - mx) * INV_GAMMA);
        sw  += e;
        swd += e * d;
      }
      const float S = swd / sw;
      Sav[p] = (mt == 0) ? S : 0.5f * (Sav[p] + S);
    }
    __builtin_amdgcn_wave_barrier();   // keep loads before next-mt stores
  }

  #pragma unroll
  for (int p = 0; p < 4; ++p) {
    const int idx = lane * 4 + p;
    const int r = idx >> 3, cl = idx & 7;
    Savg[(long)(m0 + r) * C_TOTAL + (c0 + cl)] = Sav[p];
  }
}

// ---------------------------------------------------------------------------
// Pass 2: per-row logsumexp over 10000 classes with label margin correction.
// One block per row i in [0,128).  lossv[i] = logsumexp(S_corrected) - scs.
// ---------------------------------------------------------------------------
__global__ __launch_bounds__(256) void st_pass2(const float* __restrict__ Savg,
                                                const int* __restrict__ labels,
                                                float* __restrict__ lossv) {
  __shared__ float red[256];
  const int i = blockIdx.x;
  const int t = threadIdx.x;
  const float* row = Savg + (long)i * C_TOTAL;
  const int   lab = labels[i];
  const float scs = LAMBDA_ * (row[lab] - MARGIN);   // S_correct_scaled

  float mx = -3.4e38f;
  for (int c = t; c < C_TOTAL; c += 256) {
    const float v = (c == lab) ? scs : LAMBDA_ * row[c];
    mx = fmaxf(mx, v);
  }
  red[t] = mx;
  __syncthreads();
  for (int s = 128; s > 0; s >>= 1) {
    if (t < s) red[t] = fmaxf(red[t], red[t + s]);
    __syncthreads();
  }
  mx = red[0];
  __syncthreads();

  float sum = 0.f;
  for (int c = t; c < C_TOTAL; c += 256) {
    const float v = (c == lab) ? scs : LAMBDA_ * row[c];
    sum += expf(v - mx);
  }
  red[t] = sum;
  __syncthreads();
  for (int s = 128; s > 0; s >>= 1) {
    if (t < s) red[t] += red[t + s];
    __syncthreads();
  }
  if (t == 0) {
    const float lse = mx + logf(red[0]);
    lossv[i] = lse - scs;   // = -(scs - denom)
  }
}

// ---------------------------------------------------------------------------
// Pass 3: mean of the 128 per-row losses -> scalar output.
// ---------------------------------------------------------------------------
__global__ __launch_bounds__(128) void st_pass3(const float* __restrict__ lossv,
                                                float* __restrict__ out) {
  __shared__ float red[128];
  const int t = threadIdx.x;
  red[t] = lossv[t];
  __syncthreads();
  for (int s = 64; s > 0; s >>= 1) {
    if (t < s) red[t] += red[t + s];
    __syncthreads();
  }
  if (t == 0) out[0] = red[0] * (1.0f / 128.0f);
}

extern "C" void kernel_launch(void* const* d_in, const int* in_sizes, int n_in,
                              void* d_out, int out_size, void* d_ws, size_t ws_size,
                              hipStream_t stream) {
  (void)in_sizes; (void)n_in; (void)out_size; (void)ws_size;
  const float* F      = (const float*)d_in[0];   // features (256,128) f32
  const int*   labels = (const int*)d_in[1];     // labels (256,) int
  const float* P      = (const float*)d_in[2];   // proxies (10000,10,128) f32

  float* Savg  = (float*)d_ws;                   // 128*10000 f32 = 5.12 MB
  float* lossv = Savg + (size_t)128 * C_TOTAL;   // 128 f32

  st_pass1<<<C_TOTAL / 8, 256, 0, stream>>>(F, P, Savg);
  st_pass2<<<128, 256, 0, stream>>>(Savg, labels, lossv);
  st_pass3<<<1, 128, 0, stream>>>(lossv, (float*)d_out);
}